// BallGrouping_84052509982746
// MI455X (gfx1250) — compile-verified
//
#include <hip/hip_runtime.h>

typedef __attribute__((ext_vector_type(2))) float v2f;
typedef __attribute__((ext_vector_type(4))) float v4f;
typedef __attribute__((ext_vector_type(8))) float v8f;

#define NSAMPLE 32
#define MAXR2 (0.08f * 0.08f)

// One wave32 handles 16 consecutive targets.
// Per iteration: TWO 16x16 distance tiles (32 sources) via V_WMMA_F32_16X16X4_F32,
// with the next iteration's source points preloaded (software pipeline) and the
// stream prefetched ~256 points ahead at WGP scope (near-cache fill).
//   A (16x4, M=targets): rows = (-2x_t, -2y_t, -2z_t, 1)
//   B (4x16, N=sources): cols = ( x_s ,  y_s ,  z_s , |s|^2)
//   C = |s|^2 - 2 t.s  ;  in-ball  <=>  C <= R^2 - |t|^2  (MIN_R == 0)
__global__ __launch_bounds__(256) void BallGrouping_84052509982746_kernel(
    const float* __restrict__ sp,   // [B*n, 3]
    const float* __restrict__ tp,   // [B*m, 3]
    const float* __restrict__ sf,   // [B*n, c]
    float* __restrict__ out_grouped,// [B*m, NSAMPLE, 3+c]
    float* __restrict__ out_idx,    // [B*m, NSAMPLE] (written as float)
    int n, int m, int c, int ntarget)
{
    __shared__ int idxbuf[8][16][NSAMPLE];   // per-wave slice: 16 targets x 32 idx

    const int lane = threadIdx.x & 31;
    const int wav  = threadIdx.x >> 5;
    const int waveId = blockIdx.x * (blockDim.x >> 5) + wav;
    const int tgt0 = waveId * 16;
    const bool active = (tgt0 < ntarget);

    const int k  = lane & 15;
    const bool hi = lane >= 16;

    int srcBase = 0;
    unsigned cnt = hi ? (unsigned)NSAMPLE : 0u;

    if (active) {
        const int batch = tgt0 / m;
        srcBase = batch * n;

        // ---- per-wave target setup -------------------------------------
        const int tg = tgt0 + k;
        const float tx = tp[tg * 3 + 0];
        const float ty = tp[tg * 3 + 1];
        const float tz = tp[tg * 3 + 2];
        const float tt = tx * tx + ty * ty + tz * tz;

        // A-matrix f32 16x4 layout: lanes 0-15 hold K0,K1; lanes 16-31 hold K2,K3
        v2f a;
        a.x = hi ? (-2.0f * tz) : (-2.0f * tx);
        a.y = hi ? 1.0f         : (-2.0f * ty);

        // Per-C-VGPR thresholds: lanes 0-15 need thr(row v), lanes 16-31 thr(row v+8)
        const float thbase = MAXR2 - tt;
        const int   thbits = __float_as_int(thbase);
        float thr[8];
#pragma unroll
        for (int v = 0; v < 8; ++v) {
            const float lo = __int_as_float(__builtin_amdgcn_readlane(thbits, v));
            const float h2 = __int_as_float(__builtin_amdgcn_readlane(thbits, v + 8));
            thr[v] = hi ? h2 : lo;
        }

        // 16x16 distance tile -> per-owner-lane 16-bit hit mask
        auto tile_mask = [&](float sx, float sy, float sz, unsigned& anyb) -> unsigned {
            const float ssq = sx * sx + sy * sy + sz * sz;
            // B-matrix f32 4x16 layout: VGPR0 = K0(lo)/K2(hi), VGPR1 = K1(lo)/K3(hi)
            v2f b;
            b.x = hi ? sz  : sx;
            b.y = hi ? ssq : sy;
            v8f cacc = {};
            cacc = __builtin_amdgcn_wmma_f32_16x16x4_f32(
                false, a, false, b, (short)0, cacc, false, false);
            unsigned mask = 0u;
#pragma unroll
            for (int v = 0; v < 8; ++v) {
                const unsigned bv = __builtin_amdgcn_ballot_w32(cacc[v] <= thr[v]);
                anyb |= bv;
                // bits 0-15: row v (owner lane v); bits 16-31: row v+8 (owner lane v+8)
                mask = (lane == v)     ? (bv & 0xFFFFu) : mask;
                mask = (lane == v + 8) ? (bv >> 16)     : mask;
            }
            return mask;
        };

        auto append = [&](unsigned mask, int base) {
            while (mask) {
                const int bit = __builtin_ctz(mask);
                mask &= mask - 1u;
                if (cnt < NSAMPLE) {
                    idxbuf[wav][k][cnt] = base + bit;     // batch-local source idx
                    ++cnt;
                } else {
                    break;
                }
            }
        };

        // ---- pipelined scan: 32 sources (2 WMMA tiles) per iteration ----
        const int rowA0 = (srcBase + k) * 3;
        float ax = sp[rowA0 + 0],  ay = sp[rowA0 + 1],  az = sp[rowA0 + 2];
        const int rowB0 = (srcBase + 16 + k) * 3;
        float bx = sp[rowB0 + 0],  by = sp[rowB0 + 1],  bz = sp[rowB0 + 2];

        for (int j0 = 0; j0 < n; j0 += 32) {
            // preload next iteration's 32 source points (clamped, in-range)
            const int jn = (j0 + 32 < n) ? (j0 + 32) : 0;
            const int rnA = (srcBase + jn + k) * 3;
            const float nax = sp[rnA + 0], nay = sp[rnA + 1], naz = sp[rnA + 2];
            const int rnB = (srcBase + jn + 16 + k) * 3;
            const float nbx = sp[rnB + 0], nby = sp[rnB + 1], nbz = sp[rnB + 2];

            // stream prefetch ~256 points ahead, WGP scope (fill near caches)
            const int jp = (j0 + 256 < n) ? (j0 + 256) : 0;
            __builtin_prefetch(&sp[(size_t)(srcBase + jp + lane) * 3], 0, 3);

            unsigned anyA = 0u, anyB = 0u;
            const unsigned maskA = tile_mask(ax, ay, az, anyA);
            const unsigned maskB = tile_mask(bx, by, bz, anyB);

            if (anyA) append(maskA, j0);          // tile A first: ascending idx order
            if (anyB) append(maskB, j0 + 16);

            // early exit once every target has 32 samples (hi lanes preset to 32)
            if (__builtin_amdgcn_ballot_w32(cnt >= NSAMPLE) == 0xFFFFFFFFu) break;

            ax = nax; ay = nay; az = naz;
            bx = nbx; by = nby; bz = nbz;
        }

        // ---- pad (pointops semantics: repeat first hit, or 0 if none) ---
        if (!hi) {
            const int pad = (cnt > 0u) ? idxbuf[wav][k][0] : 0;
            for (unsigned s = cnt; s < NSAMPLE; ++s) idxbuf[wav][k][s] = pad;
        }
    }

    __syncthreads();

    if (active) {
        // ---- gather + write: iteration t = target, lane = sample --------
        const int cdim = 3 + c;
        for (int t = 0; t < 16; ++t) {
            const int tgt = tgt0 + t;
            const float qx = tp[tgt * 3 + 0];
            const float qy = tp[tgt * 3 + 1];
            const float qz = tp[tgt * 3 + 2];

            const int li = idxbuf[wav][t][lane];
            const int gi = srcBase + li;

            float* orow = out_grouped + ((size_t)tgt * NSAMPLE + lane) * (size_t)cdim;
            orow[0] = sp[gi * 3 + 0] - qx;
            orow[1] = sp[gi * 3 + 1] - qy;
            orow[2] = sp[gi * 3 + 2] - qz;

            const float* frow = sf + (size_t)gi * (size_t)c;   // 128B-aligned row (c=32)
            for (int f = 0; f < c; f += 4) {
                const v4f fv = *(const v4f*)(frow + f);
                orow[3 + f + 0] = fv.x;
                orow[3 + f + 1] = fv.y;
                orow[3 + f + 2] = fv.z;
                orow[3 + f + 3] = fv.w;
            }
            out_idx[(size_t)tgt * NSAMPLE + lane] = (float)gi;
        }
    }
}

extern "C" void kernel_launch(void* const* d_in, const int* in_sizes, int n_in,
                              void* d_out, int out_size, void* d_ws, size_t ws_size,
                              hipStream_t stream) {
    const float* sf = (const float*)d_in[0];   // source_f [B*n, c]
    const float* sp = (const float*)d_in[1];   // source_p [B*n, 3]
    const float* tp = (const float*)d_in[3];   // target_p [B*m, 3]

    const int B = in_sizes[2];                 // source_o has B entries
    const int n = in_sizes[1] / (3 * B);
    const int m = in_sizes[3] / (3 * B);
    const int c = in_sizes[0] / (B * n);
    const int ntarget = B * m;

    float* out_grouped = (float*)d_out;
    float* out_idx = out_grouped + (size_t)ntarget * NSAMPLE * (size_t)(3 + c);

    const int wavesTotal    = (ntarget + 15) / 16;
    const int wavesPerBlock = 8;               // 256 threads = 8 wave32
    const int blocks = (wavesTotal + wavesPerBlock - 1) / wavesPerBlock;

    BallGrouping_84052509982746_kernel<<<blocks, 256, 0, stream>>>(
        sp, tp, sf, out_grouped, out_idx, n, m, c, ntarget);
}